// WaveRNN_72988674228445
// MI455X (gfx1250) — compile-verified
//
#include <hip/hip_runtime.h>

// =====================================================================
// 2-D elastic velocity-stress FDTD, persistent-kernel formulation for
// MI455X (gfx1250, wave32).
//   - 16 workgroups = 4 batches x 4 z-slabs of 48 rows; all 5 fields
//     LDS-resident (~212KB of the 320KB/WGP).
//   - 256 time steps inside one launch; neighbor halo exchange through
//     device-scope release/acquire flags in d_ws (no relaunch, no
//     grid-wide barrier).
//   - z-derivatives (forward & backward first differences) are banded
//     matrix products executed on the matrix pipe as chains of
//     V_WMMA_F32_16X16X4_F32 (4 per tile + one masked VALU tap for the
//     k=16 band element); x-derivatives + material math stay on VALU.
// =====================================================================

// ---------------- problem constants (match reference) ----------------
#define Bn      4
#define NTt     256
#define NZz     192
#define NXx     192
#define NRECr   128
#define DTc     0.001f
#define INV_DH  0.1f

// ---------------- decomposition ----------------
#define NSLAB    4              // z-slabs per batch -> 16 workgroups
#define SLABROWS 48
#define LROWS    50             // 48 interior rows + halo above + halo below
#define PITCH    212            // LDS pitch (floats): 212%64=20 -> WMMA-B
                                // half-wave rows (2 apart: +40 banks) and
                                // pointwise half-wave rows (8 apart: +32
                                // banks) are conflict-free
#define FSZ      (LROWS * PITCH)
#define NWAVES   12
#define NTHREADS (NWAVES * 32)
#define NTILES   36             // (48/16) tile-rows * (192/16) tile-cols

// ---------------- workspace layout (32-bit words) ----------------
#define FLAG_STRIDE 32          // one cacheline-ish per flag
#define FLAGS_WORDS (16 * 4 * FLAG_STRIDE)
#define HALO_WORDS  (16 * 4 * NXx)
#define WS_WORDS    (FLAGS_WORDS + HALO_WORDS)

enum { F_VX0 = 0, F_VZ47 = 1, F_TZZ0 = 2, F_TXZ47 = 3 };

typedef float v2f __attribute__((ext_vector_type(2)));
typedef float v8f __attribute__((ext_vector_type(8)));

__device__ __forceinline__ int* flag_ptr(int* ws, int wg, int f) {
  return ws + (wg * 4 + f) * FLAG_STRIDE;
}
__device__ __forceinline__ float* halo_ptr(int* ws, int wg, int f) {
  return (float*)(ws + FLAGS_WORDS) + (wg * 4 + f) * NXx;
}

__device__ __forceinline__ void wait_flag(int* f, int target) {
  while (__hip_atomic_load(f, __ATOMIC_ACQUIRE, __HIP_MEMORY_SCOPE_AGENT) < target)
    __builtin_amdgcn_s_sleep(1);
}
__device__ __forceinline__ void stamp_flag(int* f, int v) {
  __hip_atomic_store(f, v, __ATOMIC_RELEASE, __HIP_MEMORY_SCOPE_AGENT);
}
__device__ __forceinline__ float ld_dev(const float* p) {
  return __hip_atomic_load(p, __ATOMIC_RELAXED, __HIP_MEMORY_SCOPE_AGENT);
}
__device__ __forceinline__ void st_dev(float* p, float v) {
  __hip_atomic_store(p, v, __ATOMIC_RELAXED, __HIP_MEMORY_SCOPE_AGENT);
}

// ---------------------------------------------------------------------
// z-differences of a 16x16 tile for TWO fields on the f32 matrix pipe.
//   D[m][n] = (f[W+m+1][x0+n] - f[W+m][x0+n]) * INV_DH
// i.e. D = A @ F_window with bidiagonal A (A[m][m]=-1/DH, A[m][m+1]=+1/DH)
// over a 17-row K-window starting at LDS row W.  k=0..15 are four chained
// V_WMMA_F32_16X16X4_F32 per field; the k=16 band element touches only
// row m=15, handled as one EXEC-masked VALU FMA into element 7 of the
// upper half-wave.  All 18 LDS loads are issued before the 8 WMMAs so a
// single s_wait_dscnt covers the whole block, and the two independent
// accumulation chains interleave on the matrix pipe.
// Per-lane operand layout follows the ISA 16x4 A / 4x16 B striping:
//   A: lane m=lane&15; VGPR0/1 = K columns (2*hi, 2*hi+1)
//   B: lane n=lane&15; VGPR0/1 = K rows    (2*hi, 2*hi+1)
//   D: lane n=lane&15; VGPR v  = row v + 8*hi   (consumed by caller)
// Forward diff at tile row R0: W = R0.  Backward diff: W = R0-1.
// W+16 <= 49 always, so no clamping is needed.
// ---------------------------------------------------------------------
__device__ __forceinline__ void wmma_zdiff_pair(
    const float* fA, int WA, const float* fB, int WB, int x0,
    const v2f* aK, v8f& outA, v8f& outB)
{
  const int lane = threadIdx.x & 31;
  const int hi   = lane >> 4;
  const int col  = x0 + (lane & 15);

  float bA[8], bB[8];
#pragma unroll
  for (int c = 0; c < 4; ++c) {
    const int k0 = 4 * c + 2 * hi;
    bA[2 * c + 0] = fA[(WA + k0    ) * PITCH + col];
    bA[2 * c + 1] = fA[(WA + k0 + 1) * PITCH + col];
    bB[2 * c + 0] = fB[(WB + k0    ) * PITCH + col];
    bB[2 * c + 1] = fB[(WB + k0 + 1) * PITCH + col];
  }
  const float tailA = fA[(WA + 16) * PITCH + col];
  const float tailB = fB[(WB + 16) * PITCH + col];

  v8f accA = {0.f, 0.f, 0.f, 0.f, 0.f, 0.f, 0.f, 0.f};
  v8f accB = {0.f, 0.f, 0.f, 0.f, 0.f, 0.f, 0.f, 0.f};
#pragma unroll
  for (int c = 0; c < 4; ++c) {
    v2f ba; ba.x = bA[2 * c]; ba.y = bA[2 * c + 1];
    accA = __builtin_amdgcn_wmma_f32_16x16x4_f32(
        false, aK[c], false, ba, (short)0, accA, false, false);
    v2f bb; bb.x = bB[2 * c]; bb.y = bB[2 * c + 1];
    accB = __builtin_amdgcn_wmma_f32_16x16x4_f32(
        false, aK[c], false, bb, (short)0, accB, false, false);
  }
  if (hi) {  // k==16 band tap: D[15][n] += f[W+16][n] / DH  (m=15 -> elem 7)
    accA[7] += INV_DH * tailA;
    accB[7] += INV_DH * tailB;
  }
  outA = accA;
  outB = accB;
}

// ---------------------------------------------------------------------
__global__ void ws_init_kernel(int* ws) {
  const int i = blockIdx.x * blockDim.x + threadIdx.x;
  if (i < WS_WORDS) ws[i] = 0;
}

// ---------------------------------------------------------------------
__global__ __launch_bounds__(NTHREADS) void wave_persist_kernel(
    const float* __restrict__ xw,  const float* __restrict__ vp,
    const float* __restrict__ vs,  const float* __restrict__ rho,
    const int*   __restrict__ src_loc, const int* __restrict__ rec_loc,
    float* __restrict__ out, int* __restrict__ ws)
{
  extern __shared__ float smem[];
  float* vxs  = smem + 0 * FSZ;
  float* vzs  = smem + 1 * FSZ;
  float* txxs = smem + 2 * FSZ;
  float* tzzs = smem + 3 * FSZ;
  float* txzs = smem + 4 * FSZ;

  const int wg  = blockIdx.x;
  const int b   = wg >> 2;
  const int s   = wg & 3;
  const int tid = threadIdx.x;
  const int lane = tid & 31;
  const int wv   = tid >> 5;
  const int z0g  = s * SLABROWS;

  // zero all fields (incl. halo rows -> matches zero-pad semantics at
  // the physical top/bottom boundaries, and gives finite WMMA reads)
  for (int i = tid; i < 5 * FSZ; i += NTHREADS) smem[i] = 0.0f;
  __syncthreads();

  // loop-invariant A-band operands (bidiagonal +-1/DH), kept in VGPRs
  v2f aK[4];
  {
    const int m  = lane & 15;
    const int hi = lane >> 4;
#pragma unroll
    for (int c = 0; c < 4; ++c) {
      const int k0 = 4 * c + 2 * hi;
      aK[c].x = (k0     == m) ? -INV_DH : ((k0     == m + 1) ? INV_DH : 0.0f);
      aK[c].y = (k0 + 1 == m) ? -INV_DH : ((k0 + 1 == m + 1) ? INV_DH : 0.0f);
    }
  }

  const int sz = src_loc[2 * b + 0];
  const int sx = src_loc[2 * b + 1];

  for (int t = 0; t < NTt; ++t) {
    // ================= velocity half-step =================
    // needs txz row above (dzb) and tzz row below (dzf) from step t-1
    if (tid == 0) {
      if (s > 0)         wait_flag(flag_ptr(ws, wg - 1, F_TXZ47), t);
      if (s < NSLAB - 1) wait_flag(flag_ptr(ws, wg + 1, F_TZZ0),  t);
    }
    __syncthreads();
    if (tid < NXx) {
      if (s > 0)
        txzs[0 * PITCH + tid] = ld_dev(halo_ptr(ws, wg - 1, F_TXZ47) + tid);
      if (s < NSLAB - 1)
        tzzs[(LROWS - 1) * PITCH + tid] = ld_dev(halo_ptr(ws, wg + 1, F_TZZ0) + tid);
    }
    __syncthreads();

    for (int tile = wv; tile < NTILES; tile += NWAVES) {
      const int tr = tile / 12, tc = tile % 12;
      const int R0 = 1 + tr * 16, x0 = tc * 16;
      v8f dzb_txz, dzf_tzz;  // backward diff of txz, forward diff of tzz
      wmma_zdiff_pair(txzs, R0 - 1, tzzs, R0, x0, aK, dzb_txz, dzf_tzz);
      const int n = lane & 15, hi = lane >> 4, x = x0 + n;
#pragma unroll
      for (int v = 0; v < 8; ++v) {
        const int m  = v + 8 * hi;       // matches WMMA C/D striping
        const int R  = R0 + m;           // LDS row
        const int gz = z0g + (R - 1);    // global z
        const float txx_c   = txxs[R * PITCH + x];
        const float dxf_txx = (x < NXx - 1)
            ? (txxs[R * PITCH + x + 1] - txx_c) * INV_DH : 0.0f;
        const float txz_c   = txzs[R * PITCH + x];
        const float dxb_txz = (x > 0)
            ? (txz_c - txzs[R * PITCH + x - 1]) * INV_DH : 0.0f;
        const float dzb = (gz > 0)       ? dzb_txz[v] : 0.0f;  // pad row
        const float dzf = (gz < NZz - 1) ? dzf_tzz[v] : 0.0f;  // pad row
        const float buoy = DTc / rho[gz * NXx + x];            // L2-resident
        vxs[R * PITCH + x] += buoy * (dxf_txx + dzb);
        vzs[R * PITCH + x] += buoy * (dxb_txz + dzf);
      }
    }
    __syncthreads();

    // record receivers (out[t, r, b] = vx ; out[t, r, B+b] = vz)
    if (tid < NRECr) {
      const int rz = rec_loc[2 * tid + 0], rx = rec_loc[2 * tid + 1];
      if ((rz / SLABROWS) == s) {
        const int R = rz - z0g + 1;
        out[(t * NRECr + tid) * (2 * Bn) + b]      = vxs[R * PITCH + rx];
        out[(t * NRECr + tid) * (2 * Bn) + Bn + b] = vzs[R * PITCH + rx];
      }
    }
    // publish velocity halos: vx row0 -> slab above, vz row47 -> slab below
    if (tid < NXx) {
      if (s > 0)
        st_dev(halo_ptr(ws, wg, F_VX0) + tid, vxs[1 * PITCH + tid]);
      if (s < NSLAB - 1)
        st_dev(halo_ptr(ws, wg, F_VZ47) + tid, vzs[(LROWS - 2) * PITCH + tid]);
    }
    __threadfence();
    __syncthreads();
    if (tid == 0) {
      if (s > 0)         stamp_flag(flag_ptr(ws, wg, F_VX0),  t + 1);
      if (s < NSLAB - 1) stamp_flag(flag_ptr(ws, wg, F_VZ47), t + 1);
    }

    // ================= stress half-step =================
    // needs vz row above (dzb) and vx row below (dzf) from this step
    if (tid == 0) {
      if (s > 0)         wait_flag(flag_ptr(ws, wg - 1, F_VZ47), t + 1);
      if (s < NSLAB - 1) wait_flag(flag_ptr(ws, wg + 1, F_VX0),  t + 1);
    }
    __syncthreads();
    if (tid < NXx) {
      if (s > 0)
        vzs[0 * PITCH + tid] = ld_dev(halo_ptr(ws, wg - 1, F_VZ47) + tid);
      if (s < NSLAB - 1)
        vxs[(LROWS - 1) * PITCH + tid] = ld_dev(halo_ptr(ws, wg + 1, F_VX0) + tid);
    }
    __syncthreads();

    for (int tile = wv; tile < NTILES; tile += NWAVES) {
      const int tr = tile / 12, tc = tile % 12;
      const int R0 = 1 + tr * 16, x0 = tc * 16;
      v8f dzb_vz, dzf_vx;
      wmma_zdiff_pair(vzs, R0 - 1, vxs, R0, x0, aK, dzb_vz, dzf_vx);
      const int n = lane & 15, hi = lane >> 4, x = x0 + n;
#pragma unroll
      for (int v = 0; v < 8; ++v) {
        const int m  = v + 8 * hi;
        const int R  = R0 + m;
        const int gz = z0g + (R - 1);
        const float vx_c   = vxs[R * PITCH + x];
        const float dvxdx  = (x > 0)
            ? (vx_c - vxs[R * PITCH + x - 1]) * INV_DH : 0.0f;
        const float vz_c   = vzs[R * PITCH + x];
        const float dxf_vz = (x < NXx - 1)
            ? (vzs[R * PITCH + x + 1] - vz_c) * INV_DH : 0.0f;
        const float dvzdz   = (gz > 0)       ? dzb_vz[v] : 0.0f;
        const float dzf_vxv = (gz < NZz - 1) ? dzf_vx[v] : 0.0f;
        const int   gi  = gz * NXx + x;
        const float rv  = rho[gi], vpv = vp[gi], vsv = vs[gi];
        const float mu     = rv * vsv * vsv;
        const float lam    = rv * vpv * vpv - 2.0f * mu;
        const float lam2mu = lam + 2.0f * mu;
        txxs[R * PITCH + x] += DTc * (lam2mu * dvxdx + lam    * dvzdz);
        tzzs[R * PITCH + x] += DTc * (lam    * dvxdx + lam2mu * dvzdz);
        txzs[R * PITCH + x] += DTc * mu * (dzf_vxv + dxf_vz);
      }
    }
    __syncthreads();

    // source injection (after stress update, before halo publish)
    if (tid == 0 && (sz / SLABROWS) == s) {
      const float w = xw[b * NTt + t];
      const int R = sz - z0g + 1;
      txxs[R * PITCH + sx] += w;
      tzzs[R * PITCH + sx] += w;
    }
    __syncthreads();

    // publish stress halos: tzz row0 -> slab above, txz row47 -> slab below
    if (tid < NXx) {
      if (s > 0)
        st_dev(halo_ptr(ws, wg, F_TZZ0) + tid, tzzs[1 * PITCH + tid]);
      if (s < NSLAB - 1)
        st_dev(halo_ptr(ws, wg, F_TXZ47) + tid, txzs[(LROWS - 2) * PITCH + tid]);
    }
    __threadfence();
    __syncthreads();
    if (tid == 0) {
      if (s > 0)         stamp_flag(flag_ptr(ws, wg, F_TZZ0),  t + 1);
      if (s < NSLAB - 1) stamp_flag(flag_ptr(ws, wg, F_TXZ47), t + 1);
    }
  }
}

// ---------------------------------------------------------------------
extern "C" void kernel_launch(void* const* d_in, const int* in_sizes, int n_in,
                              void* d_out, int out_size, void* d_ws, size_t ws_size,
                              hipStream_t stream) {
  (void)in_sizes; (void)n_in; (void)out_size; (void)ws_size;
  const float* xw  = (const float*)d_in[0];   // (B, NT, 1)  f32
  const float* vp  = (const float*)d_in[1];   // (NZ, NX)    f32
  const float* vs  = (const float*)d_in[2];   // (NZ, NX)    f32
  const float* rho = (const float*)d_in[3];   // (NZ, NX)    f32
  const int* src_loc = (const int*)d_in[4];   // (B, 2)      i32
  const int* rec_loc = (const int*)d_in[5];   // (NREC, 2)   i32

  // deterministic flag/halo state every call
  ws_init_kernel<<<(WS_WORDS + 255) / 256, 256, 0, stream>>>((int*)d_ws);

  const size_t lds_bytes = (size_t)(5 * FSZ) * sizeof(float);  // ~212 KB
  wave_persist_kernel<<<Bn * NSLAB, NTHREADS, lds_bytes, stream>>>(
      xw, vp, vs, rho, src_loc, rec_loc, (float*)d_out, (int*)d_ws);
}